// ContrastiveLoss_rec3_31550829756781
// MI455X (gfx1250) — compile-verified
//
#include <hip/hip_runtime.h>

typedef __attribute__((ext_vector_type(2))) float v2f;
typedef __attribute__((ext_vector_type(8))) float v8f;

#define DD 512
#define BB 512

// ---------------------------------------------------------------------------
// Kernel 1: P[1024, 512] = [m ; tr_m] @ W^T + b  using V_WMMA_F32_16X16X4_F32
// One wave (32 lanes) computes one 16x16 output tile; K=512 -> 128 WMMA ops.
// Tiles: 64 (M) x 32 (N) = 2048 tiles; 8 waves/block -> 256 blocks.
// ---------------------------------------------------------------------------
__global__ __launch_bounds__(256) void gemm_wmma_f32(
    const float* __restrict__ m,
    const float* __restrict__ tr_m,
    const float* __restrict__ W,
    const float* __restrict__ bias,
    float* __restrict__ P) {
  const int lane = threadIdx.x & 31;
  const int wave = threadIdx.x >> 5;
  const int tile = blockIdx.x * 8 + wave;   // 0..2047
  const int tm = tile >> 5;                  // 0..63  (16 rows each, M=1024)
  const int tn = tile & 31;                  // 0..31  (16 cols each, N=512)

  const int l15   = lane & 15;
  const int khalf = (lane >> 4) * 2;         // K sub-offset per lane half

  // A tile rows: rows 0..511 come from m, 512..1023 from tr_m (tile-uniform).
  const int mrow = tm * 16 + l15;
  const float* __restrict__ arow =
      (tm < 32) ? (m + (size_t)mrow * DD)
                : (tr_m + (size_t)(mrow - BB) * DD);
  // B tile: Bmat[k, n] = W[n, k]  (pm = m @ W^T). Lane holds column n = l15.
  const float* __restrict__ wrow = W + (size_t)(tn * 16 + l15) * DD;

  v8f c = {};
#pragma unroll 4
  for (int kk = 0; kk < DD; kk += 4) {
    const int k0 = kk + khalf;
    v2f a, b;
    a.x = arow[k0];
    a.y = arow[k0 + 1];
    b.x = wrow[k0];
    b.y = wrow[k0 + 1];
    // 8 args: (neg_a, A, neg_b, B, c_mod, C, reuse_a, reuse_b)
    c = __builtin_amdgcn_wmma_f32_16x16x4_f32(
        false, a, false, b, (short)0, c, false, false);
  }

  // Bias add: every C element in this lane has N = lane&15.
  const float bv = bias[tn * 16 + l15];
#pragma unroll
  for (int v = 0; v < 8; ++v) c[v] += bv;

  // Store per documented C/D layout: element (M = v + 8*(lane>>4), N = lane&15)
  const int ncol = tn * 16 + l15;
#pragma unroll
  for (int v = 0; v < 8; ++v) {
    const int Mr = tm * 16 + v + 8 * (lane >> 4);
    P[(size_t)Mr * DD + ncol] = c[v];
  }
}

// ---------------------------------------------------------------------------
// Kernel 2: per-row diagonal dots + combine + clip -> cost[i]
// One 256-thread block per row i.
// ---------------------------------------------------------------------------
__global__ __launch_bounds__(256) void diag_cost(
    const float* __restrict__ P,
    const float* __restrict__ A_is_t,
    const float* __restrict__ A_em_t,
    float* __restrict__ cost) {
  const int i = blockIdx.x;
  const float* __restrict__ pm  = P + (size_t)i * DD;
  const float* __restrict__ ptr = P + (size_t)(i + BB) * DD;
  const float* __restrict__ ais = A_is_t + (size_t)i * (BB + 1) * DD; // A[i,i,:]
  const float* __restrict__ aem = A_em_t + (size_t)i * (BB + 1) * DD;

  float s_is = 0.f, s_em = 0.f, s_ist = 0.f, s_emt = 0.f;
  for (int d = threadIdx.x; d < DD; d += 256) {
    const float a1 = ais[d], a2 = aem[d];
    const float p1 = pm[d],  p2 = ptr[d];
    s_is  += p1 * a1;
    s_em  += p1 * a2;
    s_ist += p2 * a1;
    s_emt += p2 * a2;
  }
  // wave32 butterfly reduce
#pragma unroll
  for (int off = 16; off > 0; off >>= 1) {
    s_is  += __shfl_xor(s_is,  off, 32);
    s_em  += __shfl_xor(s_em,  off, 32);
    s_ist += __shfl_xor(s_ist, off, 32);
    s_emt += __shfl_xor(s_emt, off, 32);
  }
  __shared__ float red[8][4];
  const int lane = threadIdx.x & 31;
  const int wave = threadIdx.x >> 5;
  if (lane == 0) {
    red[wave][0] = s_is; red[wave][1] = s_em;
    red[wave][2] = s_ist; red[wave][3] = s_emt;
  }
  __syncthreads();
  if (threadIdx.x == 0) {
    float t0 = 0.f, t1 = 0.f, t2 = 0.f, t3 = 0.f;
#pragma unroll
    for (int w = 0; w < 8; ++w) {
      t0 += red[w][0]; t1 += red[w][1]; t2 += red[w][2]; t3 += red[w][3];
    }
    const float diag_is_all = 0.2f * t0 + 0.8f * t2;
    const float diag_em_all = 0.8f * t3 + 0.2f * t1;
    const float c = 0.2f + diag_is_all - diag_em_all;
    cost[i] = c > 0.f ? c : 0.f;
  }
}

// ---------------------------------------------------------------------------
// Kernel 3: deterministic tree reduction of 512 costs -> out[0]
// ---------------------------------------------------------------------------
__global__ __launch_bounds__(256) void reduce_cost(
    const float* __restrict__ cost, float* __restrict__ out) {
  __shared__ float s[256];
  const int t = threadIdx.x;
  s[t] = cost[t] + cost[t + 256];
  __syncthreads();
  for (int off = 128; off > 0; off >>= 1) {
    if (t < off) s[t] += s[t + off];
    __syncthreads();
  }
  if (t == 0) out[0] = s[0];
}

extern "C" void kernel_launch(void* const* d_in, const int* in_sizes, int n_in,
                              void* d_out, int out_size, void* d_ws, size_t ws_size,
                              hipStream_t stream) {
  const float* A_is = (const float*)d_in[0];  // [512,512,512]
  const float* A_em = (const float*)d_in[1];  // [512,512,512]
  const float* m    = (const float*)d_in[2];  // [512,512]
  const float* tr_m = (const float*)d_in[3];  // [512,512]
  const float* W    = (const float*)d_in[4];  // [512,512]
  const float* b    = (const float*)d_in[5];  // [512]
  float* out = (float*)d_out;

  float* P    = (float*)d_ws;                 // 1024*512 floats = 2 MB
  float* cost = P + (size_t)2 * BB * DD;      // 512 floats

  gemm_wmma_f32<<<256, 256, 0, stream>>>(m, tr_m, W, b, P);
  diag_cost<<<BB, 256, 0, stream>>>(P, A_is, A_em, cost);
  reduce_cost<<<1, 256, 0, stream>>>(cost, out);
}